// BSplineActivation_35802847380036
// MI455X (gfx1250) — compile-verified
//
#include <hip/hip_runtime.h>

#define NUM_KNOTS      8
#define SPLINE_ORDER   3
#define IN_FEATURES    1024
#define BATCH          8192
#define N_BASES        (NUM_KNOTS + SPLINE_ORDER - 1)     // 10
#define N_GRID         (NUM_KNOTS + 2 * SPLINE_ORDER)     // 14
#define N_INTERVALS    (N_GRID - 1)                       // 13
#define EPSV           1e-8f

#define KPAD           12     // K (=N_BASES) padded to multiple of 4 for WMMA
#define NPAD           64     // N (=13 intervals * 4 poly coeffs = 52) padded to 64
#define PP_STRIDE      52     // floats per feature in ppoly table (13 intervals * 4)

#define F_TILE         256    // features per block in the eval kernel
#define BROWS          64     // batch rows per block in the eval kernel

typedef float v2f __attribute__((ext_vector_type(2)));
typedef float v8f __attribute__((ext_vector_type(8)));

__device__ __forceinline__ float gridf(int i) {
    // jnp.linspace(-1, 1, 14): step = 2/13
    return fmaf((float)i, 2.0f / 13.0f, -1.0f);
}

// ---------------------------------------------------------------------------
// Kernel 0: build padded basis-polynomial matrix P (KPAD x NPAD).
// P[n, j*4+d] = coefficient of t^d of basis B_n restricted to interval j,
// where t = x - grid[j].  Exact Cox-de Boor recursion with the reference's
// EPS in denominators, done symbolically on cubics.  One wave, 13 live lanes.
// ---------------------------------------------------------------------------
__global__ void __launch_bounds__(32)
bspline_polybuild_kernel(float* __restrict__ Bmat) {
    const int tid = threadIdx.x;
    for (int i = tid; i < KPAD * NPAD; i += 32) Bmat[i] = 0.0f;
    __syncthreads();
    if (tid >= N_INTERVALS) return;

    const int   jj = tid;
    const float gj = gridf(jj);

    // c[i][d]: poly (in t) of basis B_{i,k} on interval jj
    float c[N_INTERVALS][4];
#pragma unroll
    for (int i = 0; i < N_INTERVALS; ++i) {
#pragma unroll
        for (int d = 0; d < 4; ++d) c[i][d] = 0.0f;
        c[i][0] = (i == jj) ? 1.0f : 0.0f;   // level-0 indicator
    }

#pragma unroll
    for (int k = 1; k <= SPLINE_ORDER; ++k) {
        float nc[N_INTERVALS][4];
#pragma unroll
        for (int i = 0; i < N_INTERVALS; ++i) {
            if (i < N_INTERVALS - k) {
                // left factor:  (x - g_i)      = (gj - g_i) + t
                // right factor: (g_{i+k+1}-x)  = (g_{i+k+1} - gj) - t
                const float la = gj - gridf(i);
                const float il = 1.0f / (gridf(i + k)     - gridf(i)     + EPSV);
                const float ra = gridf(i + k + 1) - gj;
                const float ir = 1.0f / (gridf(i + k + 1) - gridf(i + 1) + EPSV);
#pragma unroll
                for (int d = 0; d < 4; ++d) {
                    const float q = la * c[i][d]     + (d > 0 ? c[i][d - 1]     : 0.0f);
                    const float s = ra * c[i + 1][d] - (d > 0 ? c[i + 1][d - 1] : 0.0f);
                    nc[i][d] = q * il + s * ir;
                }
            }
        }
#pragma unroll
        for (int i = 0; i < N_INTERVALS; ++i)
            if (i < N_INTERVALS - k)
#pragma unroll
                for (int d = 0; d < 4; ++d) c[i][d] = nc[i][d];
    }

#pragma unroll
    for (int n = 0; n < N_BASES; ++n)
#pragma unroll
        for (int d = 0; d < 4; ++d)
            Bmat[n * NPAD + jj * 4 + d] = c[n][d];
}

// ---------------------------------------------------------------------------
// Kernel 1: ppoly = coeff (1024 x 10) * P (10 x 52) via fp32 WMMA 16x16x4.
// One wave per 16x16 output tile; grid = 64 M-tiles x 4 N-tiles.
// A fragment (16x4 f32): lane L -> row M = L%16; VGPR0 holds K = (L<16?0:2),
// VGPR1 holds K = (L<16?1:3).  B fragment (4x16): VGPR r holds rows r / r+2
// striped across lane halves.  C/D: 8 VGPRs per ISA 7.12.2.
// ---------------------------------------------------------------------------
__global__ void __launch_bounds__(32)
bspline_wmma_ppoly_kernel(const float* __restrict__ coeff,
                          const float* __restrict__ Bmat,
                          float* __restrict__ ppoly) {
    const int lane = threadIdx.x;
    const int half = lane >> 4;       // 0: lanes 0-15, 1: lanes 16-31
    const int l16  = lane & 15;
    const int mt   = blockIdx.x >> 2; // 64 tiles over features
    const int nt   = blockIdx.x & 3;  // 4 tiles over (interval,degree)
    const int m    = mt * 16 + l16;

    v8f acc = {};
#pragma unroll
    for (int kk = 0; kk < KPAD; kk += 4) {
        const int ka = kk + 2 * half;
        v2f a, b;
        a.x = (ka     < N_BASES) ? coeff[m * N_BASES + ka]     : 0.0f;
        a.y = (ka + 1 < N_BASES) ? coeff[m * N_BASES + ka + 1] : 0.0f;
        b.x = Bmat[(ka    ) * NPAD + nt * 16 + l16];
        b.y = Bmat[(ka + 1) * NPAD + nt * 16 + l16];
        acc = __builtin_amdgcn_wmma_f32_16x16x4_f32(
            /*neg_a=*/false, a, /*neg_b=*/false, b,
            /*c_mod=*/(short)0, acc, /*reuse_a=*/false, /*reuse_b=*/false);
    }

#pragma unroll
    for (int r = 0; r < 8; ++r) {
        const int f = mt * 16 + r + half * 8;
        const int n = nt * 16 + l16;
        if (n < PP_STRIDE) ppoly[f * PP_STRIDE + n] = acc[r];
    }
}

// ---------------------------------------------------------------------------
// Kernel 2: streaming eval.  out[b,f] = Horner(ppoly[f, j(x)], t) with
// t = clamp(x) - grid[j].  j > 12 (x == 1.0 after clamp) -> 0, matching the
// reference's half-open indicator at the right boundary.
// 256 threads: 64 feature-quads x 4 row-lanes; float4 global traffic;
// poly table staged in LDS (208 B/feature -> aligned ds_load_b128 lookups).
// ---------------------------------------------------------------------------
__global__ void __launch_bounds__(256)
bspline_eval_kernel(const float* __restrict__ x,
                    const float* __restrict__ ppoly,
                    float* __restrict__ out) {
    __shared__ float lds[F_TILE * PP_STRIDE];   // 53248 bytes

    const int tid   = threadIdx.x;
    const int ftile = blockIdx.x & (IN_FEATURES / F_TILE - 1);  // 0..3
    const int btile = blockIdx.x >> 2;
    const int fbase = ftile * F_TILE;

    // Stage this block's slice of the poly table into LDS (coalesced float4).
    const float4* src = (const float4*)(ppoly + (size_t)fbase * PP_STRIDE);
    float4*       dst = (float4*)lds;
#pragma unroll
    for (int i = tid; i < F_TILE * PP_STRIDE / 4; i += 256) dst[i] = src[i];
    __syncthreads();

    const int fq = tid & 63;   // which feature quad (4 consecutive features)
    const int rl = tid >> 6;   // row lane 0..3
    const int b0 = btile * BROWS;

    for (int b = b0 + rl; b < b0 + BROWS; b += 4) {
        const float* xr = x + (size_t)b * IN_FEATURES + fbase + fq * 4;
        if (b + 4 < b0 + BROWS)
            __builtin_prefetch(xr + 4 * IN_FEATURES, 0, 1);  // global_prefetch_b8

        const float4 xv = *(const float4*)xr;
        const float xs[4] = {xv.x, xv.y, xv.z, xv.w};
        float os[4];
#pragma unroll
        for (int c = 0; c < 4; ++c) {
            const float xc = fminf(fmaxf(xs[c], -1.0f), 1.0f);
            const float s  = (xc + 1.0f) * 6.5f;        // 6.5 = 13/2 = 1/h
            const int   j  = (int)s;
            const int   jc = j > 12 ? 12 : j;
            const float t  = xc - fmaf((float)jc, 2.0f / 13.0f, -1.0f);
            const float4 p = *(const float4*)&lds[(fq * 4 + c) * PP_STRIDE + jc * 4];
            const float o  = fmaf(fmaf(fmaf(p.w, t, p.z), t, p.y), t, p.x);
            os[c] = (j > 12) ? 0.0f : o;   // x == 1.0 -> empty indicator -> 0
        }
        float4 ov;
        ov.x = os[0]; ov.y = os[1]; ov.z = os[2]; ov.w = os[3];
        *(float4*)(out + (size_t)b * IN_FEATURES + fbase + fq * 4) = ov;
    }
}

extern "C" void kernel_launch(void* const* d_in, const int* in_sizes, int n_in,
                              void* d_out, int out_size, void* d_ws, size_t ws_size,
                              hipStream_t stream) {
    const float* x     = (const float*)d_in[0];   // (8192, 1024) f32
    const float* coeff = (const float*)d_in[1];   // (1024, 10)  f32
    float*       out   = (float*)d_out;           // (8192, 1024) f32

    float* Bmat  = (float*)d_ws;                          // 12*64 f32 = 3 KB
    float* ppoly = (float*)((char*)d_ws + 4096);          // 1024*52 f32 = 208 KB

    bspline_polybuild_kernel<<<1, 32, 0, stream>>>(Bmat);
    bspline_wmma_ppoly_kernel<<<(IN_FEATURES / 16) * (NPAD / 16), 32, 0, stream>>>(
        coeff, Bmat, ppoly);
    bspline_eval_kernel<<<(IN_FEATURES / F_TILE) * (BATCH / BROWS), 256, 0, stream>>>(
        x, ppoly, out);
}